// SphericalBasisLayer_49366354100415
// MI455X (gfx1250) — compile-verified
//
#include <hip/hip_runtime.h>
#include <math.h>

#define NSPH 7
#define NRAD 6
#define NC   42              /* NSPH*NRAD */
#define LUT_N 96             /* 42 zeros | 42 norm | 7 ycoef | pad */
#define LUT_F 256            /* float offset of rbf table inside d_ws   */
#define BLOCK 256
#define NROW 32              /* row slots per block in gather kernel */

// ---------------------------------------------------------------------------
// Host-side table construction — mirrors the reference numpy code exactly
// (double-precision bisection for spherical Bessel zeros, then f32 cast).
// Recomputed every call (deterministic), shipped as by-value kernel args.
// ---------------------------------------------------------------------------
namespace {

double jn_h(double z, int n) {
    double j0 = sin(z) / z;
    if (n == 0) return j0;
    double j1 = sin(z) / (z * z) - cos(z) / z;
    double jm1 = j0, jc = j1;
    for (int m = 1; m < n; ++m) {
        double t = (2.0 * m + 1.0) / z * jc - jm1;
        jm1 = jc; jc = t;
    }
    return jc;
}

double bisect_h(int nn, double a, double b) {
    double fa = jn_h(a, nn);
    for (int it = 0; it < 100; ++it) {
        double m = 0.5 * (a + b);
        double fm = jn_h(m, nn);
        if (fa * fm <= 0.0) { b = m; }
        else                { a = m; fa = fm; }
    }
    return 0.5 * (a + b);
}

void build_lut(float* lut) {
    const int n = NSPH, k = NRAD;
    const double PI = 3.14159265358979323846;
    double zerosj[NSPH][NRAD];
    double pts[NSPH + NRAD - 1];
    double rac[NSPH + NRAD - 1];

    for (int j = 0; j < k; ++j) zerosj[0][j] = (j + 1) * PI;
    for (int j = 0; j < k + n - 1; ++j) pts[j] = (j + 1) * PI;
    for (int j = 0; j < k + n - 1; ++j) rac[j] = 0.0;

    for (int i = 1; i < n; ++i) {
        int cnt = k + n - 1 - i;
        for (int j = 0; j < cnt; ++j)
            rac[j] = bisect_h(i, pts[j], pts[j + 1]);
        for (int j = 0; j < k + n - 1; ++j) pts[j] = rac[j];
        for (int j = 0; j < k; ++j) zerosj[i][j] = rac[j];
    }

    for (int l = 0; l < n; ++l)
        for (int i = 0; i < k; ++i) {
            lut[l * k + i] = (float)zerosj[l][i];
            double jj = jn_h(zerosj[l][i], l + 1);
            lut[NC + l * k + i] = (float)(1.0 / sqrt(0.5 * jj * jj));
        }
    for (int l = 0; l < n; ++l)
        lut[2 * NC + l] = (float)sqrt((2.0 * l + 1.0) / (4.0 * PI));
    for (int i = 2 * NC + n; i < LUT_N; ++i) lut[i] = 0.0f;
}

} // namespace

struct TableArg { float v[LUT_N]; };

__global__ void write_lut_kernel(TableArg ta, float* __restrict__ dst) {
    unsigned i = threadIdx.x;
    if (i < LUT_N) dst[i] = ta.v[i];
}

// ---------------------------------------------------------------------------
// Kernel A: per-edge radial basis table  rbf[e, c] = env(x)*NORM[c]*j_l(Z[c]*x)
// Two elements (even pair) per thread; pairs always share l and alignment.
// LUT staged into LDS via CDNA5 async global->LDS loads.
// ---------------------------------------------------------------------------
__global__ __launch_bounds__(BLOCK) void rbf_kernel(
    const float* __restrict__ dist,
    const float* __restrict__ lutg,
    float*       __restrict__ rbf,
    unsigned halfN)                       // = E*42/2
{
    __shared__ float s_lut[LUT_N];
    const unsigned tid = threadIdx.x;

    if (tid < LUT_N) {
        unsigned lds_off = (unsigned)(size_t)&s_lut[tid];
        unsigned goff    = tid * 4u;
        asm volatile("global_load_async_to_lds_b32 %0, %1, %2"
                     :
                     : "v"(lds_off), "v"(goff), "s"((unsigned long long)lutg)
                     : "memory");
    }
    asm volatile("s_wait_asynccnt 0" ::: "memory");
    __syncthreads();

    unsigned gid = blockIdx.x * BLOCK + tid;
    if (gid >= halfN) return;
    unsigned ja = gid * 2u;
    unsigned e  = ja / 42u;
    unsigned c  = ja - e * 42u;           // even
    unsigned l  = c / 6u;                 // same l for c and c+1

    const float x  = dist[e] * 0.2f;      // dist / CUTOFF
    // envelope, exact reference association: 1/x + a*x^5 + b*x^6 + c*x^6*x
    float x2  = x * x;
    float x4  = x2 * x2;
    float xp  = x4 * x;                   // x^5
    float xp1 = xp * x;                   // x^6
    float env = 1.0f / x + (-21.0f) * xp + 35.0f * xp1 + (-15.0f) * xp1 * x;
    env = (x < 1.0f) ? env : 0.0f;

    float z0 = s_lut[c] * x;
    float z1 = s_lut[c + 1u] * x;
    float s0 = sinf(z0), c0 = cosf(z0);
    float s1 = sinf(z1), c1 = cosf(z1);
    float r0 = 1.0f / z0, r1 = 1.0f / z1;
    float a0 = s0 * r0, a1 = s1 * r1;     // j0
    if (l > 0u) {
        float b0 = (a0 - c0) * r0;        // j1 = sin/z^2 - cos/z
        float b1 = (a1 - c1) * r1;
        float p0 = a0, p1 = a1;
        for (unsigned m = 1u; m < l; ++m) {
            float k  = 2.0f * (float)m + 1.0f;
            float n0 = k * r0 * b0 - p0;
            float n1 = k * r1 * b1 - p1;
            p0 = b0; p1 = b1; b0 = n0; b1 = n1;
        }
        a0 = b0; a1 = b1;
    }
    float2 o;
    o.x = env * (s_lut[NC + c] * a0);
    o.y = env * (s_lut[NC + c + 1u] * a1);
    *(float2*)&rbf[ja] = o;
}

// ---------------------------------------------------------------------------
// Kernel B: out[t, c] = rbf[idx_kj[t], c] * (YC[l] * P_l(cos(angle[t])))
// 4 elements per thread (two even pairs; a quad may straddle a row boundary,
// each pair cannot).  32 threads/block precompute the <=26 distinct rows'
// cbf into LDS once; element threads do gather-multiply-store (b64 loads,
// b128 stores).
// ---------------------------------------------------------------------------
__global__ __launch_bounds__(BLOCK) void gather_kernel(
    const float* __restrict__ angle,
    const int*   __restrict__ idx_kj,
    const float* __restrict__ lutg,
    const float* __restrict__ rbf,
    float*       __restrict__ out,
    unsigned total, unsigned T)
{
    __shared__ float s_cbf[NROW * NSPH];
    __shared__ int   s_e[NROW];
    const unsigned tid  = threadIdx.x;
    const unsigned blk0 = blockIdx.x * (BLOCK * 4u);     // first element index
    const unsigned t_first = blk0 / 42u;

    if (tid < NROW) {
        unsigned t    = t_first + tid;
        unsigned t_hi = (blk0 + (BLOCK * 4u - 1u)) / 42u;
        if (t <= t_hi && t < T) {
            s_e[tid] = idx_kj[t];
            float ct = cosf(angle[t]);
            float P0 = 1.0f, P1 = ct;
            float P2 = ( 3.0f * ct * P1 -        P0) / 2.0f;
            float P3 = ( 5.0f * ct * P2 - 2.0f * P1) / 3.0f;
            float P4 = ( 7.0f * ct * P3 - 3.0f * P2) / 4.0f;
            float P5 = ( 9.0f * ct * P4 - 4.0f * P3) / 5.0f;
            float P6 = (11.0f * ct * P5 - 5.0f * P4) / 6.0f;
            s_cbf[tid * NSPH + 0] = lutg[2 * NC + 0] * P0;
            s_cbf[tid * NSPH + 1] = lutg[2 * NC + 1] * P1;
            s_cbf[tid * NSPH + 2] = lutg[2 * NC + 2] * P2;
            s_cbf[tid * NSPH + 3] = lutg[2 * NC + 3] * P3;
            s_cbf[tid * NSPH + 4] = lutg[2 * NC + 4] * P4;
            s_cbf[tid * NSPH + 5] = lutg[2 * NC + 5] * P5;
            s_cbf[tid * NSPH + 6] = lutg[2 * NC + 6] * P6;
        }
        if (tid == 0u) {                                  // stream-ahead prefetch
            unsigned tp = t_first + 8192u;
            if (tp >= T) tp = T - 1u;
            __builtin_prefetch(&angle[tp], 0, 0);
            __builtin_prefetch(&idx_kj[tp], 0, 0);
        }
    }
    __syncthreads();

    unsigned j4 = blk0 + tid * 4u;
    if (j4 >= total) return;

    // pair 0: elements (j4, j4+1)
    unsigned t0 = j4 / 42u;
    unsigned c0 = j4 - t0 * 42u;          // even
    unsigned l0 = c0 / 6u;
    unsigned r0 = t0 - t_first;
    int      e0 = s_e[r0];
    float    w0 = s_cbf[r0 * NSPH + l0];
    const float2 a = *(const float2*)&rbf[(unsigned)e0 * 42u + c0];

    if (j4 + 3u < total) {
        // pair 1: elements (j4+2, j4+3) — may be in the next row
        unsigned j2 = j4 + 2u;
        unsigned t1 = j2 / 42u;
        unsigned c1 = j2 - t1 * 42u;      // even
        unsigned l1 = c1 / 6u;
        unsigned r1 = t1 - t_first;
        int      e1 = s_e[r1];
        float    w1 = s_cbf[r1 * NSPH + l1];
        const float2 b = *(const float2*)&rbf[(unsigned)e1 * 42u + c1];

        float4 o;
        o.x = a.x * w0; o.y = a.y * w0;
        o.z = b.x * w1; o.w = b.y * w1;
        *(float4*)&out[j4] = o;           // 16B aligned (j4 % 4 == 0)
    } else {
        // tail (total % 4 == 2): single pair
        float2 o; o.x = a.x * w0; o.y = a.y * w0;
        *(float2*)&out[j4] = o;
    }
}

// ---------------------------------------------------------------------------
// Fallback (ws too small for the rbf table): monolithic per-element kernel.
// ---------------------------------------------------------------------------
__global__ __launch_bounds__(BLOCK) void sbl_fallback_kernel(
    const float* __restrict__ dist,
    const float* __restrict__ angle,
    const int*   __restrict__ idx_kj,
    const float* __restrict__ lutg,
    float*       __restrict__ out,
    unsigned total)
{
    __shared__ float s_lut[LUT_N];
    const unsigned tid = threadIdx.x;
    if (tid < LUT_N) {
        unsigned lds_off = (unsigned)(size_t)&s_lut[tid];
        unsigned goff    = tid * 4u;
        asm volatile("global_load_async_to_lds_b32 %0, %1, %2"
                     :: "v"(lds_off), "v"(goff), "s"((unsigned long long)lutg)
                     : "memory");
    }
    asm volatile("s_wait_asynccnt 0" ::: "memory");
    __syncthreads();

    unsigned j = blockIdx.x * BLOCK + tid;
    if (j >= total) return;
    unsigned t = j / 42u, c = j % 42u, l = c / 6u;

    const int   e = idx_kj[t];
    const float x = dist[e] * 0.2f;
    float x2 = x * x, x4 = x2 * x2, xp = x4 * x, xp1 = xp * x;
    float env = 1.0f / x + (-21.0f) * xp + 35.0f * xp1 + (-15.0f) * xp1 * x;
    env = (x < 1.0f) ? env : 0.0f;

    float z = s_lut[c] * x;
    float sz = sinf(z), cz = cosf(z);
    float rz = 1.0f / z;
    float jl = sz * rz;
    if (l > 0u) {
        float jm1 = jl, jc = (jl - cz) * rz;
        for (unsigned m = 1u; m < l; ++m) {
            float jx = (2.0f * (float)m + 1.0f) * rz * jc - jm1;
            jm1 = jc; jc = jx;
        }
        jl = jc;
    }
    float rbf = env * (s_lut[NC + c] * jl);

    float ct = cosf(angle[t]);
    float P = 1.0f;
    if (l > 0u) {
        float Pm1 = 1.0f, Pc = ct;
        for (unsigned m = 1u; m < l; ++m) {
            float Pn = ((2.0f * (float)m + 1.0f) * ct * Pc - (float)m * Pm1)
                       / ((float)m + 1.0f);
            Pm1 = Pc; Pc = Pn;
        }
        P = Pc;
    }
    out[j] = rbf * (s_lut[2 * NC + l] * P);
}

// ---------------------------------------------------------------------------
extern "C" void kernel_launch(void* const* d_in, const int* in_sizes, int n_in,
                              void* d_out, int out_size, void* d_ws, size_t ws_size,
                              hipStream_t stream) {
    const float* dist   = (const float*)d_in[0];
    const float* angle  = (const float*)d_in[1];
    const int*   idx_kj = (const int*)  d_in[2];
    float*       out    = (float*)d_out;

    const unsigned E = (unsigned)in_sizes[0];
    const unsigned T = (unsigned)in_sizes[1];
    const unsigned total = T * (unsigned)NC;

    TableArg ta;
    build_lut(ta.v);

    float* lut_dev = (float*)d_ws;
    write_lut_kernel<<<1, 128, 0, stream>>>(ta, lut_dev);

    const unsigned rbfN = E * (unsigned)NC;
    const size_t   need = (size_t)(LUT_F + rbfN) * sizeof(float);

    if (ws_size >= need) {
        float* rbf_dev = (float*)d_ws + LUT_F;
        unsigned halfA  = rbfN / 2u;
        unsigned quadsB = (total + 3u) / 4u;
        rbf_kernel<<<dim3((halfA + BLOCK - 1) / BLOCK), dim3(BLOCK), 0, stream>>>(
            dist, lut_dev, rbf_dev, halfA);
        gather_kernel<<<dim3((quadsB + BLOCK - 1) / BLOCK), dim3(BLOCK), 0, stream>>>(
            angle, idx_kj, lut_dev, rbf_dev, out, total, T);
    } else {
        unsigned blocks = (total + BLOCK - 1) / BLOCK;
        sbl_fallback_kernel<<<dim3(blocks), dim3(BLOCK), 0, stream>>>(
            dist, angle, idx_kj, lut_dev, out, total);
    }
}